// Net_GCN_3994319585976
// MI455X (gfx1250) — compile-verified
//
#include <hip/hip_runtime.h>

// ---- problem dims ----
#define B_    32
#define T_    512
#define G_    2000
#define PPG_  10
#define P_    20000
#define E_    64000
#define NOUT_ 10
#define KDIM  (G_ * 2)        // 4000

// ---- tiling for the streaming einsum ----
#define TSPLIT 8
#define TCH    (T_ / TSPLIT)  // 64
#define BT     4
#define NBT    (B_ / BT)      // 8

// ---- GEMM K-split ----
#define KCHUNKS 8
#define KCH     (KDIM / KCHUNKS)   // 500 (multiple of 4)

typedef float v2f __attribute__((ext_vector_type(2)));
typedef float v8f __attribute__((ext_vector_type(8)));

// ---- workspace layout (float element offsets) ----
#define WS_PARTIAL 0
#define WS_H       (TSPLIT * B_ * G_)      // 512000
#define WS_S       (WS_H + B_ * G_)        // 576000
#define WS_DEG     (WS_S + B_ * G_)        // 640000
#define WS_DINV    (WS_DEG + G_)           // 642000
#define WS_ACT     (WS_DINV + G_)          // 644000
#define WS_WPAD    (WS_ACT + B_ * KDIM)    // 772000  (16 x 4000 padded W_out)

__global__ void k_zero(float* __restrict__ p, int n) {
  int i = blockIdx.x * blockDim.x + threadIdx.x;
  if (i < n) p[i] = 0.0f;
}

// Stage 1: per_peak einsum fused with per-gene peak reduction, split over
// t-chunks and batch tiles. Bandwidth-bound: streams x (1.31 GB) once;
// W_sub (41 MB) is L2-resident and amortized 4x via the batch tile.
__global__ void k_perpeak(const float* __restrict__ x,
                          const float* __restrict__ Wsub,
                          float* __restrict__ partial) {
  int tid = blockIdx.x * blockDim.x + threadIdx.x;
  if (tid >= G_ * TSPLIT * NBT) return;
  int g     = tid % G_;          // adjacent threads -> adjacent genes (dense p span)
  int rem   = tid / G_;
  int chunk = rem % TSPLIT;
  int btile = rem / TSPLIT;
  int b0    = btile * BT;

  const long pbase = (long)g * PPG_;
  float acc[BT] = {0.f, 0.f, 0.f, 0.f};

  int t0 = chunk * TCH;
  for (int t = t0; t < t0 + TCH; ++t) {
    const float* wrow = Wsub + (long)t * P_ + pbase;
    const float* xrow = x + ((long)b0 * T_ + (long)t) * P_ + pbase;
#pragma unroll
    for (int q = 0; q < PPG_ / 2; ++q) {
      v2f w2 = *(const v2f*)(wrow + 2 * q);      // 8B-aligned (40g + 8q)
#pragma unroll
      for (int bb = 0; bb < BT; ++bb) {
        v2f x2 = *(const v2f*)(xrow + (long)bb * T_ * P_ + 2 * q);
        acc[bb] += x2.x * w2.x + x2.y * w2.y;
      }
    }
  }
#pragma unroll
  for (int bb = 0; bb < BT; ++bb)
    partial[((long)chunk * B_ + (b0 + bb)) * G_ + g] = acc[bb];
}

// Stage 2: reduce t-chunk partials, add bias, ReLU -> h[b,g]
__global__ void k_reduce_h(const float* __restrict__ partial,
                           const float* __restrict__ b_sub,
                           float* __restrict__ h) {
  int tid = blockIdx.x * blockDim.x + threadIdx.x;
  if (tid >= B_ * G_) return;
  int g = tid % G_, b = tid / G_;
  float s = b_sub[g];
#pragma unroll
  for (int c = 0; c < TSPLIT; ++c)
    s += partial[((long)c * B_ + b) * G_ + g];
  h[(long)b * G_ + g] = fmaxf(s, 0.0f);
}

// Degree count: +1.0f atomics; integer-valued fp sums are exact -> deterministic
__global__ void k_deg(const int* __restrict__ col, float* __restrict__ deg) {
  int e = blockIdx.x * blockDim.x + threadIdx.x;
  if (e < E_) atomicAdd(&deg[col[e]], 1.0f);
}

__global__ void k_dinv(const float* __restrict__ deg, float* __restrict__ dinv) {
  int g = blockIdx.x * blockDim.x + threadIdx.x;
  if (g < G_) {
    float d = deg[g];
    dinv[g] = (d > 0.0f) ? (1.0f / sqrtf(d)) : 0.0f;
  }
}

// GCN message scatter: s[b,col] += norm * h[b,row]  (GCNConv(1->2) factorizes)
__global__ void k_scatter(const int* __restrict__ row, const int* __restrict__ col,
                          const float* __restrict__ dinv, const float* __restrict__ h,
                          float* __restrict__ s) {
  int tid = blockIdx.x * blockDim.x + threadIdx.x;
  if (tid >= E_ * NBT) return;
  int e  = tid % E_;
  int bq = tid / E_;
  int r = row[e], c = col[e];
  float nrm = dinv[r] * dinv[c];
#pragma unroll
  for (int bb = 0; bb < BT; ++bb) {
    int b = bq * BT + bb;
    atomicAdd(&s[(long)b * G_ + c], nrm * h[(long)b * G_ + r]);
  }
}

// Apply W_gcn / b_gcn channels + ReLU -> act[b, 2g+c] (GEMM A operand, fp32)
__global__ void k_act(const float* __restrict__ s, const float* __restrict__ Wgcn,
                      const float* __restrict__ bgcn, float* __restrict__ act) {
  int tid = blockIdx.x * blockDim.x + threadIdx.x;
  if (tid >= B_ * G_) return;
  int g = tid % G_, b = tid / G_;
  float v = s[(long)b * G_ + g];
  act[(long)b * KDIM + 2 * g]     = fmaxf(v * Wgcn[0] + bgcn[0], 0.0f);
  act[(long)b * KDIM + 2 * g + 1] = fmaxf(v * Wgcn[1] + bgcn[1], 0.0f);
}

// Zero-pad W_out [10,4000] -> Wpad [16,4000] so the WMMA B-operand loads are
// unconditional (no exec-masked loads in the GEMM inner loop).
__global__ void k_pad_w(const float* __restrict__ Wout, float* __restrict__ Wpad) {
  int tid = blockIdx.x * blockDim.x + threadIdx.x;
  if (tid >= 16 * KDIM) return;
  int n = tid / KDIM;
  Wpad[tid] = (n < NOUT_) ? Wout[tid] : 0.0f;
}

// Output GEMM out[32,10] = act[32,4000] @ Wpad^T + b_out via
// V_WMMA_F32_16X16X4_F32. One 512-thread block = 16 waves:
// wave w -> (M-tile = w&1, K-chunk = w>>1); 125 WMMA steps per wave.
// Partial C fragments reduced through LDS in a fixed order (deterministic).
__global__ void __launch_bounds__(512)
k_out_wmma(const float* __restrict__ act, const float* __restrict__ Wpad,
           const float* __restrict__ bout, float* __restrict__ out) {
  __shared__ float red[16 * 256];   // 16 waves x (8 regs x 32 lanes)

  int lane = threadIdx.x & 31;
  int w    = threadIdx.x >> 5;
  int tile = w & 1;                  // M tile: rows 16*tile .. 16*tile+15
  int kc   = w >> 1;                 // K chunk
  int half = lane >> 4;
  int l    = lane & 15;
  int m    = tile * 16 + l;          // A row this lane feeds
  int n    = l;                      // B column this lane feeds (Wpad row)
  int koff = half * 2;               // per-lane K offset within a K=4 step

  const float* arow = act + (long)m * KDIM;
  const float* brow = Wpad + (long)n * KDIM;
  v8f c = {};

  int k0 = kc * KCH;
  for (int k = k0; k < k0 + KCH; k += 4) {
    v2f a = *(const v2f*)(arow + k + koff);   // 8B aligned: (16000m + 4k + 8*half)
    v2f b = *(const v2f*)(brow + k + koff);
    c = __builtin_amdgcn_wmma_f32_16x16x4_f32(false, a, false, b,
                                              (short)0, c, false, false);
  }

  // spill partial fragment: red[w][r][lane]
#pragma unroll
  for (int r = 0; r < 8; ++r)
    red[w * 256 + r * 32 + lane] = c[r];
  __syncthreads();

  // 320 threads gather the 32x10 output, summing K-chunks in fixed order.
  int tid = threadIdx.x;
  if (tid < B_ * NOUT_) {
    int mo = tid / NOUT_;
    int no = tid % NOUT_;
    int t  = mo >> 4;                 // tile
    int ml = mo & 15;
    int r  = ml & 7;
    int hf = ml >> 3;
    int ln = hf * 16 + no;
    float sum = bout[no];
#pragma unroll
    for (int q = 0; q < KCHUNKS; ++q)
      sum += red[(q * 2 + t) * 256 + r * 32 + ln];
    out[mo * NOUT_ + no] = sum;
  }
}

extern "C" void kernel_launch(void* const* d_in, const int* in_sizes, int n_in,
                              void* d_out, int out_size, void* d_ws, size_t ws_size,
                              hipStream_t stream) {
  (void)in_sizes; (void)n_in; (void)out_size; (void)ws_size;
  const float* x    = (const float*)d_in[0];
  const int*   ei   = (const int*)d_in[1];
  const float* Wsub = (const float*)d_in[2];
  const float* bsub = (const float*)d_in[3];
  const float* Wgcn = (const float*)d_in[4];
  const float* bgcn = (const float*)d_in[5];
  const float* Wout = (const float*)d_in[6];
  const float* bout = (const float*)d_in[7];
  float* out = (float*)d_out;
  float* ws  = (float*)d_ws;

  float* partial = ws + WS_PARTIAL;
  float* h       = ws + WS_H;
  float* s       = ws + WS_S;
  float* deg     = ws + WS_DEG;
  float* dinv    = ws + WS_DINV;
  float* act     = ws + WS_ACT;
  float* Wpad    = ws + WS_WPAD;
  const int* row = ei;
  const int* col = ei + E_;

  {  // zero the accumulators (s and deg are contiguous)
    int n = B_ * G_ + G_;
    k_zero<<<(n + 255) / 256, 256, 0, stream>>>(s, n);
  }
  {  // pad W_out (independent of the pipeline, can go early)
    int n = 16 * KDIM;
    k_pad_w<<<(n + 255) / 256, 256, 0, stream>>>(Wout, Wpad);
  }
  {
    int n = G_ * TSPLIT * NBT;  // 128000 threads streaming x
    k_perpeak<<<(n + 255) / 256, 256, 0, stream>>>(x, Wsub, partial);
  }
  {
    int n = B_ * G_;
    k_reduce_h<<<(n + 255) / 256, 256, 0, stream>>>(partial, bsub, h);
  }
  k_deg<<<(E_ + 255) / 256, 256, 0, stream>>>(col, deg);
  k_dinv<<<(G_ + 255) / 256, 256, 0, stream>>>(deg, dinv);
  {
    int n = E_ * NBT;
    k_scatter<<<(n + 255) / 256, 256, 0, stream>>>(row, col, dinv, h, s);
  }
  {
    int n = B_ * G_;
    k_act<<<(n + 255) / 256, 256, 0, stream>>>(s, Wgcn, bgcn, act);
  }
  k_out_wmma<<<1, 512, 0, stream>>>(act, Wpad, bout, out);
}